// Net_NN_11123965296881
// MI455X (gfx1250) — compile-verified
//
#include <hip/hip_runtime.h>
#include <math.h>

#define N_NODES 50000
#define F_IN    3703
#define F_HID   16
#define F_OUT   6
#define N_TILES (N_NODES / 16)             // 3125, exact
#define KCHUNK  128
#define NCHUNK  ((F_IN + KCHUNK - 1) / KCHUNK)   // 29
#define CHUNK_ELEMS (KCHUNK * F_HID)       // 2048 floats = 8 KB

typedef __attribute__((ext_vector_type(2)))  float        v2f;
typedef __attribute__((ext_vector_type(8)))  float        v8f;
typedef __attribute__((ext_vector_type(16))) __bf16       v16bf;
typedef __attribute__((ext_vector_type(4)))  unsigned int v4u;
typedef __attribute__((ext_vector_type(8)))  int          v8i;
typedef __attribute__((ext_vector_type(4)))  int          v4i;

// ---------------- degree / norm ----------------

__global__ void k_init_deg(float* __restrict__ deg) {
    int i = blockIdx.x * blockDim.x + threadIdx.x;
    if (i < N_NODES) deg[i] = 1.0f;   // self-loop contribution
}

__global__ void k_deg_scatter(const int* __restrict__ dst, int E,
                              float* __restrict__ deg) {
    int e = blockIdx.x * blockDim.x + threadIdx.x;
    if (e < E) atomicAdd(&deg[dst[e]], 1.0f);
}

__global__ void k_dinv(const float* __restrict__ deg, float* __restrict__ dinv) {
    int i = blockIdx.x * blockDim.x + threadIdx.x;
    if (i < N_NODES) dinv[i] = rsqrtf(deg[i]);   // deg >= 1 always
}

// ---------------- TDM: DMA one W1 K-chunk (<=8KB contiguous) into LDS --------

__device__ __forceinline__ void tdm_load_w1_chunk(const float* __restrict__ W1,
                                                  int chunk, unsigned lds_off) {
    const unsigned total     = (unsigned)F_IN * F_HID;
    const unsigned base_elem = (unsigned)chunk * CHUNK_ELEMS;
    const unsigned remain    = total - base_elem;                 // OOB guard
    const unsigned elems     = remain < CHUNK_ELEMS ? remain : CHUNK_ELEMS;
    const unsigned long long gaddr =
        (unsigned long long)(uintptr_t)(W1 + base_elem);

    // D# group 0: count=1 | lds_addr | global_addr[56:0] | type=2
    v4u g0;
    g0[0] = 1u;                                    // count=1, user mode
    g0[1] = lds_off;                               // LDS byte address
    g0[2] = (unsigned)(gaddr & 0xFFFFFFFFu);       // global_addr[31:0]
    g0[3] = (unsigned)((gaddr >> 32) & 0x1FFFFFFu) // global_addr[56:32]
          | (2u << 30);                            // type=2 ("image")

    // D# group 1: data_size=4B, 1-row tile of `elems` elements
    v8i g1;
    g1[0] = (int)(2u << 16);                       // workgroup_mask=0, data_size=2(4B)
    g1[1] = (int)((remain & 0xFFFFu) << 16);       // tensor_dim0[15:0]  (atomic_bar=0)
    g1[2] = (int)(((remain >> 16) & 0xFFFFu)       // tensor_dim0[31:16]
          | (1u << 16));                           // tensor_dim1 = 1
    g1[3] = (int)((elems & 0xFFFFu) << 16);        // tile_dim0 = elems (dim1 hi = 0)
    g1[4] = 1;                                     // tile_dim1 = 1, tile_dim2 = 0
    g1[5] = (int)remain;                           // tensor_dim0_stride lo32
    g1[6] = 0;                                     // stride hi16 | dim1_stride lo16
    g1[7] = 0;                                     // tensor_dim1_stride hi32

    v4i g2 = {0, 0, 0, 0};
    v4i g3 = {0, 0, 0, 0};
#if defined(__clang_major__) && (__clang_major__ >= 23)
    v8i g4 = {0, 0, 0, 0, 0, 0, 0, 0};
    __builtin_amdgcn_tensor_load_to_lds(g0, g1, g2, g3, g4, 0);
#else
    __builtin_amdgcn_tensor_load_to_lds(g0, g1, g2, g3, 0);
#endif
}

// ---------------- layer-1 GEMM: h1 = x @ W1 via f32 WMMA ----------------
// 8 waves/block, one 16x16 output tile per wave. W1 K-chunks are DMA'd into
// LDS by the TDM (double-buffered, overlapped with compute), B fed from LDS.

__global__ __launch_bounds__(256)
void k_gemm1_wmma(const float* __restrict__ x,
                  const float* __restrict__ W1,
                  float* __restrict__ h1) {
    __shared__ float sW[2][CHUNK_ELEMS];           // 2 x 8 KB double buffer

    const int wave = threadIdx.x >> 5;
    const int lane = threadIdx.x & 31;
    const int tile = blockIdx.x * 8 + wave;
    const bool active = (tile < N_TILES);          // inactive waves still barrier
    const int half = lane >> 4;                    // 0 or 1
    const int lr   = lane & 15;
    const float* xrow = x + (size_t)((active ? tile : 0) * 16 + lr) * F_IN;

    const unsigned lds0 = (unsigned)(uintptr_t)&sW[0][0];
    const unsigned lds1 = (unsigned)(uintptr_t)&sW[1][0];

    v8f c = {0.f,0.f,0.f,0.f,0.f,0.f,0.f,0.f};

    if (wave == 0 && lane == 0) tdm_load_w1_chunk(W1, 0, lds0);

    for (int ch = 0; ch < NCHUNK; ++ch) {
        const int cur = ch & 1;
        if (wave == 0) __builtin_amdgcn_s_wait_tensorcnt(0);  // chunk `ch` landed
        __syncthreads();                                      // publish buffer; also
                                                              // fences reads of buf
                                                              // before next DMA hits it
        if (wave == 0 && lane == 0 && ch + 1 < NCHUNK)
            tdm_load_w1_chunk(W1, ch + 1, cur ? lds0 : lds1);

        if (active) {
            const int kbase = ch * KCHUNK;
            const float* sw = &sW[cur][0];
            if (ch < NCHUNK - 1) {
                #pragma unroll 4
                for (int kk = 0; kk < KCHUNK; kk += 4) {
                    const int kl = kk + 2 * half;   // this lane-half: K = kl, kl+1
                    v2f a, b;
                    a.x = xrow[kbase + kl];
                    a.y = xrow[kbase + kl + 1];
                    b.x = sw[kl * F_HID + lr];
                    b.y = sw[(kl + 1) * F_HID + lr];
                    c = __builtin_amdgcn_wmma_f32_16x16x4_f32(false, a, false, b,
                                                              (short)0, c, false, false);
                }
            } else {
                // K tail chunk (119 valid): guard A with zeros; B stale-LDS values
                // pair only with A==0 so they contribute nothing.
                for (int kk = 0; kk < KCHUNK; kk += 4) {
                    const int kl = kk + 2 * half;
                    const int ka = kbase + kl;
                    v2f a, b;
                    a.x = (ka     < F_IN) ? xrow[ka]     : 0.f;
                    a.y = (ka + 1 < F_IN) ? xrow[ka + 1] : 0.f;
                    b.x = sw[kl * F_HID + lr];
                    b.y = sw[(kl + 1) * F_HID + lr];
                    c = __builtin_amdgcn_wmma_f32_16x16x4_f32(false, a, false, b,
                                                              (short)0, c, false, false);
                }
            }
        }
    }

    if (active) {
        // D layout: VGPR r -> M = r + 8*half, N = lane%16
        float* out = h1 + (size_t)tile * 16 * F_HID;
        #pragma unroll
        for (int r = 0; r < 8; ++r)
            out[(r + 8 * half) * F_HID + lr] = c[r];
    }
}

// ---------------- aggregation helpers ----------------

template <int F>
__global__ void k_self_init(const float* __restrict__ h,
                            const float* __restrict__ dinv,
                            float* __restrict__ agg) {
    int idx = blockIdx.x * blockDim.x + threadIdx.x;
    if (idx < N_NODES * F) {
        int i = idx / F;
        float di = dinv[i];
        agg[idx] = h[idx] * di * di;             // self-loop edge, norm = dinv^2
    }
}

__global__ void k_edge_scatter16(const int* __restrict__ src,
                                 const int* __restrict__ dst, int E,
                                 const float* __restrict__ dinv,
                                 const float* __restrict__ h,
                                 float* __restrict__ agg) {
    long long idx = (long long)blockIdx.x * blockDim.x + threadIdx.x;
    int e = (int)(idx >> 4);
    int f = (int)(idx & 15);
    if (e < E) {
        int s = src[e], d = dst[e];
        float w = dinv[s] * dinv[d];
        atomicAdd(&agg[(size_t)d * 16 + f], h[(size_t)s * 16 + f] * w);
    }
}

__global__ void k_edge_scatter6(const int* __restrict__ src,
                                const int* __restrict__ dst, int E,
                                const float* __restrict__ dinv,
                                const float* __restrict__ h,
                                float* __restrict__ agg) {
    long long idx = (long long)blockIdx.x * blockDim.x + threadIdx.x;
    int e = (int)(idx >> 3);
    int f = (int)(idx & 7);
    if (e < E && f < F_OUT) {
        int s = src[e], d = dst[e];
        float w = dinv[s] * dinv[d];
        atomicAdd(&agg[(size_t)d * F_OUT + f], h[(size_t)s * F_OUT + f] * w);
    }
}

__global__ void k_bias_relu16(const float* __restrict__ agg,
                              const float* __restrict__ b1,
                              float* __restrict__ hout) {
    int idx = blockIdx.x * blockDim.x + threadIdx.x;
    if (idx < N_NODES * F_HID) {
        float v = agg[idx] + b1[idx & 15];
        hout[idx] = v > 0.f ? v : 0.f;
    }
}

// ---------------- layer-2 GEMM (tiny: K=16, N=6) ----------------

__global__ void k_gemm2(const float* __restrict__ h,
                        const float* __restrict__ W2,
                        float* __restrict__ h2) {
    int i = blockIdx.x * blockDim.x + threadIdx.x;
    if (i >= N_NODES) return;
    const float* hr = h + (size_t)i * F_HID;
    float acc[F_OUT] = {0.f, 0.f, 0.f, 0.f, 0.f, 0.f};
    #pragma unroll
    for (int k = 0; k < F_HID; ++k) {
        float hv = hr[k];
        #pragma unroll
        for (int c = 0; c < F_OUT; ++c)
            acc[c] = fmaf(hv, W2[k * F_OUT + c], acc[c]);
    }
    #pragma unroll
    for (int c = 0; c < F_OUT; ++c)
        h2[(size_t)i * F_OUT + c] = acc[c];
}

__global__ void k_bias_logsoftmax(const float* __restrict__ agg,
                                  const float* __restrict__ b2,
                                  float* __restrict__ out) {
    int i = blockIdx.x * blockDim.x + threadIdx.x;
    if (i >= N_NODES) return;
    float o[F_OUT];
    float m = -INFINITY;
    #pragma unroll
    for (int c = 0; c < F_OUT; ++c) {
        o[c] = agg[(size_t)i * F_OUT + c] + b2[c];
        m = fmaxf(m, o[c]);
    }
    float s = 0.f;
    #pragma unroll
    for (int c = 0; c < F_OUT; ++c) s += __expf(o[c] - m);
    float ls = __logf(s);
    #pragma unroll
    for (int c = 0; c < F_OUT; ++c)
        out[(size_t)i * F_OUT + c] = o[c] - m - ls;
}

// ---------------- launcher ----------------

extern "C" void kernel_launch(void* const* d_in, const int* in_sizes, int n_in,
                              void* d_out, int out_size, void* d_ws, size_t ws_size,
                              hipStream_t stream) {
    const float* x   = (const float*)d_in[0];
    const int*   ei  = (const int*)d_in[1];       // [2, E] row-major
    const float* W1  = (const float*)d_in[2];
    const float* b1  = (const float*)d_in[3];
    const float* W2  = (const float*)d_in[4];
    const float* b2  = (const float*)d_in[5];
    float*       out = (float*)d_out;

    const int E = in_sizes[1] / 2;
    const int* src = ei;
    const int* dst = ei + E;

    // workspace layout (floats)
    float* ws   = (float*)d_ws;
    float* deg  = ws;                      // 50000
    float* dinv = deg  + N_NODES;          // 50000
    float* h1   = dinv + N_NODES;          // 50000*16
    float* agg1 = h1   + (size_t)N_NODES * F_HID;   // 50000*16
    float* h2   = agg1 + (size_t)N_NODES * F_HID;   // 50000*6
    float* agg2 = h2   + (size_t)N_NODES * F_OUT;   // 50000*6

    const int T = 256;
    // normalization
    k_init_deg   <<<(N_NODES + T - 1) / T, T, 0, stream>>>(deg);
    k_deg_scatter<<<(E + T - 1) / T,       T, 0, stream>>>(dst, E, deg);
    k_dinv       <<<(N_NODES + T - 1) / T, T, 0, stream>>>(deg, dinv);

    // layer 1: WMMA GEMM, 8 waves (8 tiles) per 256-thread block, TDM-fed LDS
    k_gemm1_wmma<<<(N_TILES + 7) / 8, 256, 0, stream>>>(x, W1, h1);

    k_self_init<F_HID><<<(N_NODES * F_HID + T - 1) / T, T, 0, stream>>>(h1, dinv, agg1);
    {
        long long tot = (long long)E * 16;
        k_edge_scatter16<<<(unsigned)((tot + T - 1) / T), T, 0, stream>>>(src, dst, E, dinv, h1, agg1);
    }
    k_bias_relu16<<<(N_NODES * F_HID + T - 1) / T, T, 0, stream>>>(agg1, b1, h1); // reuse h1

    // layer 2
    k_gemm2<<<(N_NODES + T - 1) / T, T, 0, stream>>>(h1, W2, h2);
    k_self_init<F_OUT><<<(N_NODES * F_OUT + T - 1) / T, T, 0, stream>>>(h2, dinv, agg2);
    {
        long long tot = (long long)E * 8;
        k_edge_scatter6<<<(unsigned)((tot + T - 1) / T), T, 0, stream>>>(src, dst, E, dinv, h2, agg2);
    }
    k_bias_logsoftmax<<<(N_NODES + T - 1) / T, T, 0, stream>>>(agg2, b2, out);
    (void)n_in; (void)out_size; (void)ws_size;
}